// Attention_59236188947256
// MI455X (gfx1250) — compile-verified
//
#include <hip/hip_runtime.h>
#include <hip/hip_bf16.h>
#include <math.h>

typedef float v2f __attribute__((ext_vector_type(2)));
typedef float v8f __attribute__((ext_vector_type(8)));

#define LRELU_ALPHA 0.2f
#define MASK_FILL   -1e9f

// ---------------------------------------------------------------------------
// Kernel 1: rank-2 projection via V_WMMA_F32_16X16X4_F32.
// One wave (32 threads) per 16-row tile of the flattened (B*N, F) matrix X.
//   A tile : 16x4 slice of X     (lane l: row = l&15, K = (l>>4)*2 + v)
//   B tile : 4x16, col0=w1, col1=w2, other cols 0
//   D      : 16x16 f32; D[:,0] = s1 rows, D[:,1] = s2 rows
// B is built branch-free: b = c1*w1 + c2*w2 with per-lane 0/1 constants, so
// the K-loop is straight-line (loads + FMA + wmma), EXEC all-ones throughout.
// ---------------------------------------------------------------------------
__global__ __launch_bounds__(32)
void gat_proj_wmma(const float* __restrict__ x,
                   const float* __restrict__ w1,
                   const float* __restrict__ w2,
                   float* __restrict__ s1,
                   float* __restrict__ s2,
                   int F)
{
    const int lane = threadIdx.x;            // 0..31, full wave
    const int tile = blockIdx.x;
    const int row0 = tile * 16;
    const int m    = lane & 15;              // A row / B-D column index
    const int koff = (lane >> 4) * 2;        // K sub-offset for this half-wave

    const float c1 = (m == 0) ? 1.0f : 0.0f; // selects w1 into B column 0
    const float c2 = (m == 1) ? 1.0f : 0.0f; // selects w2 into B column 1

    const float* __restrict__ xrow = x + (size_t)(row0 + m) * (size_t)F;

    v8f c = {};
    for (int k = 0; k < F; k += 4) {
        const int kk = k + koff;             // even -> 8B aligned float2 loads
        const float2 av  = *(const float2*)(xrow + kk);
        const float2 w1v = *(const float2*)(w1 + kk);
        const float2 w2v = *(const float2*)(w2 + kk);

        v2f a; a.x = av.x; a.y = av.y;
        v2f b;
        b.x = fmaf(c1, w1v.x, c2 * w2v.x);
        b.y = fmaf(c1, w1v.y, c2 * w2v.y);

        c = __builtin_amdgcn_wmma_f32_16x16x4_f32(
                /*neg_a=*/false, a, /*neg_b=*/false, b,
                /*c_mod=*/(short)0, c, /*reuse_a=*/false, /*reuse_b=*/false);
    }

    // D layout: lane l holds column n = l&15, rows (l>>4)*8 + r in VGPR r.
    const int rbase = (lane >> 4) * 8;
    if (m == 0) {
        #pragma unroll
        for (int r = 0; r < 8; ++r) s1[row0 + rbase + r] = c[r];
    } else if (m == 1) {
        #pragma unroll
        for (int r = 0; r < 8; ++r) s2[row0 + rbase + r] = c[r];
    }
}

// ---------------------------------------------------------------------------
// Kernel 2: fused outer-sum + LeakyReLU + masked-fill + row softmax.
// One 256-thread block (8 wave32) per (b, i) row of length N = 4096.
// Each thread owns 16 elements as 4 x float4 -> b128 loads/stores.
// Bandwidth-bound: 512 MB out + 128 MB mask => ~28 us floor at 23.3 TB/s.
// ---------------------------------------------------------------------------
__device__ __forceinline__ float wave_max32(float v) {
    #pragma unroll
    for (int o = 16; o > 0; o >>= 1) v = fmaxf(v, __shfl_xor(v, o, 32));
    return v;
}
__device__ __forceinline__ float wave_sum32(float v) {
    #pragma unroll
    for (int o = 16; o > 0; o >>= 1) v += __shfl_xor(v, o, 32);
    return v;
}

__global__ __launch_bounds__(256)
void gat_softmax_row(const float* __restrict__ s1,
                     const float* __restrict__ s2,
                     const unsigned char* __restrict__ mask,
                     float* __restrict__ out,
                     int N)
{
    const int i = blockIdx.x;                // row within batch
    const int b = blockIdx.y;                // batch
    const int t = threadIdx.x;               // 0..255

    const size_t rowbase = ((size_t)b * (size_t)N + (size_t)i) * (size_t)N;
    const float  s1v  = s1[(size_t)b * N + i];
    const float* s2r  = s2 + (size_t)b * N;
    const unsigned char* mrow = mask + rowbase;

    float4 e[4];
    float lmax = -INFINITY;

    #pragma unroll
    for (int g = 0; g < 4; ++g) {
        const int j = g * 1024 + t * 4;      // 16B aligned
        float4 sv = *(const float4*)(s2r + j);
        uchar4 mv = *(const uchar4*)(mrow + j);

        float4 v;
        v.x = s1v + sv.x; v.y = s1v + sv.y; v.z = s1v + sv.z; v.w = s1v + sv.w;
        v.x = (v.x >= 0.f) ? v.x : LRELU_ALPHA * v.x;
        v.y = (v.y >= 0.f) ? v.y : LRELU_ALPHA * v.y;
        v.z = (v.z >= 0.f) ? v.z : LRELU_ALPHA * v.z;
        v.w = (v.w >= 0.f) ? v.w : LRELU_ALPHA * v.w;
        v.x = mv.x ? MASK_FILL : v.x;
        v.y = mv.y ? MASK_FILL : v.y;
        v.z = mv.z ? MASK_FILL : v.z;
        v.w = mv.w ? MASK_FILL : v.w;
        e[g] = v;
        lmax = fmaxf(lmax, fmaxf(fmaxf(v.x, v.y), fmaxf(v.z, v.w)));
    }

    __shared__ float red_max[8];
    __shared__ float red_sum[8];
    const int wave = t >> 5, lane = t & 31;

    lmax = wave_max32(lmax);
    if (lane == 0) red_max[wave] = lmax;
    __syncthreads();
    float bmax = red_max[0];
    #pragma unroll
    for (int w = 1; w < 8; ++w) bmax = fmaxf(bmax, red_max[w]);

    float lsum = 0.0f;
    #pragma unroll
    for (int g = 0; g < 4; ++g) {
        float4 v = e[g];
        v.x = __expf(v.x - bmax);
        v.y = __expf(v.y - bmax);
        v.z = __expf(v.z - bmax);
        v.w = __expf(v.w - bmax);
        e[g] = v;
        lsum += (v.x + v.y) + (v.z + v.w);
    }

    lsum = wave_sum32(lsum);
    if (lane == 0) red_sum[wave] = lsum;
    __syncthreads();
    float bsum = red_sum[0];
    #pragma unroll
    for (int w = 1; w < 8; ++w) bsum += red_sum[w];

    const float inv = 1.0f / bsum;
    #pragma unroll
    for (int g = 0; g < 4; ++g) {
        const int j = g * 1024 + t * 4;
        float4 v = e[g];
        v.x *= inv; v.y *= inv; v.z *= inv; v.w *= inv;
        *(float4*)(out + rowbase + j) = v;   // global_store_b128
    }
}

// ---------------------------------------------------------------------------
// Launch: inputs are [x (B,N,F) f32], [mask (B,N,N) bool/1B], [w1 (F)], [w2 (F)]
// ---------------------------------------------------------------------------
extern "C" void kernel_launch(void* const* d_in, const int* in_sizes, int n_in,
                              void* d_out, int out_size, void* d_ws, size_t ws_size,
                              hipStream_t stream)
{
    const float*         x    = (const float*)d_in[0];
    const unsigned char* mask = (const unsigned char*)d_in[1];
    const float*         w1   = (const float*)d_in[2];
    const float*         w2   = (const float*)d_in[3];
    float*               out  = (float*)d_out;

    const long long F  = in_sizes[2];                 // 256
    const long long BN = (long long)in_sizes[0] / F;  // B*N = 32768
    const long long N  = (long long)in_sizes[1] / BN; // 4096
    const int       B  = (int)(BN / N);               // 8

    float* s1 = (float*)d_ws;                         // BN floats
    float* s2 = s1 + BN;                              // BN floats (256 KB total)

    gat_proj_wmma<<<dim3((unsigned)(BN / 16)), dim3(32), 0, stream>>>(
        x, w1, w2, s1, s2, (int)F);

    gat_softmax_row<<<dim3((unsigned)N, (unsigned)B), dim3(256), 0, stream>>>(
        s1, s2, mask, out, (int)N);
}